// SWTLayer_45045617000705
// MI455X (gfx1250) — compile-verified
//
#include <hip/hip_runtime.h>

// ---------------------------------------------------------------------------
// SWT layer on MI455X (gfx1250), wave32 + V_WMMA_F32_16X16X4_F32 everywhere.
//
// All six stages are f32 WMMA GEMMs:
//   F (256x64)  : forward Haar-SWT (norm=True) restricted to kept bands
//                 {cA6, cD6, cD5, cD1}  (truncation zeroes cD4..cD2)
//   G (64x256)  : inverse SWT with the truncated bands implicitly zero
//   per-loc mix : (32x16)x(16x16) GEMM against W[:,:,b,c]
//   skip        : x @ bias^T  (feeds pass 3b as the WMMA C operand)
// F and G are built on-device by running the reference recursions on unit
// vectors, and are stored PAIR-INTERLEAVED (Fp[k/2][r][k&1]) so each WMMA
// A-fragment is a single aligned global_load_b64.
// ---------------------------------------------------------------------------

typedef float v2f __attribute__((ext_vector_type(2)));
typedef float v8f __attribute__((ext_vector_type(8)));

__device__ __forceinline__ v8f wmma4(v2f a, v2f b, v8f c) {
  // D(16x16,f32) = A(16x4,f32) * B(4x16,f32) + C
  return __builtin_amdgcn_wmma_f32_16x16x4_f32(
      /*neg_a=*/false, a, /*neg_b=*/false, b,
      /*c_mod=*/(short)0, c, /*reuse_a=*/false, /*reuse_b=*/false);
}

// ---------------------------------------------------------------------------
// Init: Fp[kp=0..31][r=0..255][2] with Fp[kp][r][p] = F[r][2*kp+p]
//       Gp[kp=0..127][r=0..63][2] with Gp[kp][r][p] = G[r][2*kp+p]
// Thread 0..63   : forward transform of unit vector e_k  -> F[:,k]
// Thread 64..319 : inverse transform of unit band basis  -> G[:,cb]
// ---------------------------------------------------------------------------
__global__ void swt_init_mats(float* __restrict__ Fp, float* __restrict__ Gp) {
  int t = threadIdx.x;
  if (t < 64) {
    float a[64], na[64], d1[64], d5[64], d6[64];
    for (int i = 0; i < 64; ++i) a[i] = (i == t) ? 1.f : 0.f;
    for (int j = 0; j < 6; ++j) {
      int s = 1 << j;
      for (int i = 0; i < 64; ++i) {
        float x0 = a[i], x1 = a[(i + s) & 63];
        float d = (x0 - x1) * 0.5f;
        na[i] = (x0 + x1) * 0.5f;
        if (j == 0) d1[i] = d;   // cD_1 (finest, kept)
        if (j == 4) d5[i] = d;   // cD_5 (kept)
        if (j == 5) d6[i] = d;   // cD_6 (kept)
      }
      for (int i = 0; i < 64; ++i) a[i] = na[i];
    }
    float* dst = Fp + (t >> 1) * 512 + (t & 1);   // pair-interleaved column k=t
    for (int i = 0; i < 64; ++i) {
      dst[(  0 + i) * 2] = a[i];   // band 0: cA6
      dst[( 64 + i) * 2] = d6[i];  // band 1: cD6
      dst[(128 + i) * 2] = d5[i];  // band 2: cD5
      dst[(192 + i) * 2] = d1[i];  // band 3: cD1
    }
  } else if (t < 320) {
    int u = t - 64, kk = u >> 6, ii = u & 63;
    float a[64], na[64];
    for (int i = 0; i < 64; ++i) a[i] = (kk == 0 && i == ii) ? 1.f : 0.f;
    for (int j = 6; j >= 1; --j) {
      int s = 1 << (j - 1);
      int cdk = (j == 6) ? 1 : (j == 5) ? 2 : (j == 1) ? 3 : -1;
      for (int i = 0; i < 64; ++i) {
        int ip = (i - s) & 63;
        float cd0 = (cdk == kk && i  == ii) ? 1.f : 0.f;
        float cdp = (cdk == kk && ip == ii) ? 1.f : 0.f;
        na[i] = 0.5f * ((a[i] + cd0) + (a[ip] - cdp));
      }
      for (int i = 0; i < 64; ++i) a[i] = na[i];
    }
    float* dst = Gp + (u >> 1) * 128 + (u & 1);   // pair-interleaved column cb=u
    for (int i = 0; i < 64; ++i) dst[i * 2] = a[i];
  }
}

// ---------------------------------------------------------------------------
// Pass 1a: forward along c.  T1[a][b][cc][d] = sum_c F[cc,c] * X[a,b,c,d]
// ---------------------------------------------------------------------------
__global__ void __launch_bounds__(256)
swt_fwd_c(const float* __restrict__ X, const float* __restrict__ Fp,
          float* __restrict__ T1) {
  int w  = blockIdx.x * 8 + (threadIdx.x >> 5);
  int l  = threadIdx.x & 31, m = l & 15, hi = l >> 4;
  int rt = w & 15;          // cc tile
  int ab = w >> 4;          // a*64 + b
  int r0 = rt * 16;
  const float* xb = X + ab * 1024;                 // [c=64][d=16]
  const float* ap = Fp + (r0 + m) * 2 + hi * 512;  // kp = 2k + hi
  v8f acc = {0.f, 0.f, 0.f, 0.f, 0.f, 0.f, 0.f, 0.f};
  #pragma unroll 4
  for (int k = 0; k < 16; ++k) {
    int ks = 4 * k + 2 * hi;
    v2f av = *(const v2f*)(ap + k * 1024);         // A[m, ks..ks+1], one b64
    v2f bv;
    bv.x = xb[ks * 16 + m];
    bv.y = xb[(ks + 1) * 16 + m];
    acc = wmma4(av, bv, acc);
  }
  float* ob = T1 + ab * 4096;                      // [cc=256][d=16]
  #pragma unroll
  for (int v = 0; v < 8; ++v) ob[(r0 + v + 8 * hi) * 16 + m] = acc[v];
}

// ---------------------------------------------------------------------------
// Pass 1b: forward along b.  XT[cb*256+cc][a*16+d] = sum_b F[cb,b]*T1[a,b,cc,d]
// ---------------------------------------------------------------------------
__global__ void __launch_bounds__(256)
swt_fwd_b(const float* __restrict__ T1, const float* __restrict__ Fp,
          float* __restrict__ XT) {
  int w  = blockIdx.x * 8 + (threadIdx.x >> 5);
  int l  = threadIdx.x & 31, m = l & 15, hi = l >> 4;
  int rt = w & 15;
  int rest = w >> 4;
  int cc = rest & 255, a = rest >> 8;
  int r0 = rt * 16;
  const float* ap = Fp + (r0 + m) * 2 + hi * 512;
  const float* tb = T1 + a * 262144 + cc * 16 + m; // + ks*4096
  v8f acc = {0.f, 0.f, 0.f, 0.f, 0.f, 0.f, 0.f, 0.f};
  #pragma unroll 4
  for (int k = 0; k < 16; ++k) {
    int ks = 4 * k + 2 * hi;
    v2f av = *(const v2f*)(ap + k * 1024);
    v2f bv;
    bv.x = tb[ks * 4096];
    bv.y = tb[(ks + 1) * 4096];
    acc = wmma4(av, bv, acc);
  }
  #pragma unroll
  for (int v = 0; v < 8; ++v) {
    int cb = r0 + v + 8 * hi;
    XT[(cb * 256 + cc) * 512 + a * 16 + m] = acc[v];
  }
}

// ---------------------------------------------------------------------------
// Per-location channel mix (einsum 'abcd,debc->abce'), IN PLACE on XT.
// One wave owns one location: Y[a,e] = sum_d XT[a,d] * W[d,e,b,c].
// W gathered 64B-granular; the 192MB L2 keeps the 67MB live region resident.
// ---------------------------------------------------------------------------
__global__ void __launch_bounds__(256)
swt_mix(float* __restrict__ XT, const float* __restrict__ W) {
  int w  = blockIdx.x * 8 + (threadIdx.x >> 5);    // compact location
  int l  = threadIdx.x & 31, m = l & 15, hi = l >> 4;
  int cb = w >> 8, cc = w & 255;
  int bo = (cb < 192) ? cb : cb + 192;             // compact -> original band idx
  int co = (cc < 192) ? cc : cc + 192;
  float* blk = XT + w * 512;                       // [a=32][d=16]
  v2f afr[2][4];
  #pragma unroll
  for (int rt = 0; rt < 2; ++rt)
    #pragma unroll
    for (int k = 0; k < 4; ++k) {
      int ks = 4 * k + 2 * hi;
      afr[rt][k] = *(const v2f*)(blk + (rt * 16 + m) * 16 + ks);
    }
  v8f acc0 = {0.f, 0.f, 0.f, 0.f, 0.f, 0.f, 0.f, 0.f};
  v8f acc1 = {0.f, 0.f, 0.f, 0.f, 0.f, 0.f, 0.f, 0.f};
  const float* wp = W + bo * 448 + co + m * 200704;
  #pragma unroll
  for (int k = 0; k < 4; ++k) {
    int ks = 4 * k + 2 * hi;
    v2f bv;
    bv.x = wp[ks * 3211264];          // W[d=ks  ,e=m,b,c]  (16*200704 stride)
    bv.y = wp[(ks + 1) * 3211264];    // W[d=ks+1,e=m,b,c]
    acc0 = wmma4(afr[0][k], bv, acc0);
    acc1 = wmma4(afr[1][k], bv, acc1);
  }
  #pragma unroll
  for (int v = 0; v < 8; ++v) {
    int r = v + 8 * hi;
    blk[r * 16 + m]        = acc0[v];              // Y layout [a][e]
    blk[(16 + r) * 16 + m] = acc1[v];
  }
}

// ---------------------------------------------------------------------------
// Skip path: OUT[abc, e] = sum_d X[abc, d] * bias[e, d]   (x @ bias^T)
// ---------------------------------------------------------------------------
__global__ void __launch_bounds__(256)
swt_skip(const float* __restrict__ X, const float* __restrict__ Bias,
         float* __restrict__ OUT) {
  int w  = blockIdx.x * 8 + (threadIdx.x >> 5);
  int l  = threadIdx.x & 31, m = l & 15, hi = l >> 4;
  int r0 = w * 16;
  v8f acc = {0.f, 0.f, 0.f, 0.f, 0.f, 0.f, 0.f, 0.f};
  #pragma unroll
  for (int k = 0; k < 4; ++k) {
    int ks = 4 * k + 2 * hi;
    v2f av = *(const v2f*)(X + (r0 + m) * 16 + ks);
    v2f bv = *(const v2f*)(Bias + m * 16 + ks);    // B[K=d][N=e] = bias[e][d]
    acc = wmma4(av, bv, acc);
  }
  #pragma unroll
  for (int v = 0; v < 8; ++v) OUT[(r0 + v + 8 * hi) * 16 + m] = acc[v];
}

// ---------------------------------------------------------------------------
// Pass 3a: inverse along b. T2[b][cc][a*16+e] = sum_cb G[b,cb]*Y[cb,cc,a,e]
// ---------------------------------------------------------------------------
__global__ void __launch_bounds__(256)
swt_inv_b(const float* __restrict__ YT, const float* __restrict__ Gp,
          float* __restrict__ T2) {
  int w  = blockIdx.x * 8 + (threadIdx.x >> 5);
  int l  = threadIdx.x & 31, m = l & 15, hi = l >> 4;
  int rt = w & 3;
  int rest = w >> 2;
  int cc = rest & 255, a = rest >> 8;
  int r0 = rt * 16;
  const float* ap = Gp + (r0 + m) * 2 + hi * 128;  // kp = 2k + hi
  const float* yb = YT + cc * 512 + a * 16 + m;    // + ks*131072
  v8f acc = {0.f, 0.f, 0.f, 0.f, 0.f, 0.f, 0.f, 0.f};
  #pragma unroll 4
  for (int k = 0; k < 64; ++k) {
    int ks = 4 * k + 2 * hi;
    v2f av = *(const v2f*)(ap + k * 256);          // G[r0+m, ks..ks+1], one b64
    v2f bv;
    bv.x = yb[ks * 131072];
    bv.y = yb[(ks + 1) * 131072];
    acc = wmma4(av, bv, acc);
  }
  #pragma unroll
  for (int v = 0; v < 8; ++v) {
    int b = r0 + v + 8 * hi;
    T2[(b * 256 + cc) * 512 + a * 16 + m] = acc[v];
  }
}

// ---------------------------------------------------------------------------
// Pass 3b: inverse along c, accumulating onto the skip result already in OUT.
// OUT[a,b,c,e] = skip + sum_cc G[c,cc] * T2[b,cc,a,e]
// ---------------------------------------------------------------------------
__global__ void __launch_bounds__(256)
swt_inv_c(const float* __restrict__ T2, const float* __restrict__ Gp,
          float* __restrict__ OUT) {
  int w  = blockIdx.x * 8 + (threadIdx.x >> 5);
  int l  = threadIdx.x & 31, m = l & 15, hi = l >> 4;
  int rt = w & 3;
  int rest = w >> 2;
  int b = rest & 63, a = rest >> 6;
  int r0 = rt * 16;
  int obase = (a * 64 + b) * 1024;
  const float* ap = Gp + (r0 + m) * 2 + hi * 128;
  const float* tb = T2 + b * 131072 + a * 16 + m;  // + ks*512
  v8f acc;
  #pragma unroll
  for (int v = 0; v < 8; ++v)                      // C = skip values
    acc[v] = OUT[obase + (r0 + v + 8 * hi) * 16 + m];
  #pragma unroll 4
  for (int k = 0; k < 64; ++k) {
    int ks = 4 * k + 2 * hi;
    v2f av = *(const v2f*)(ap + k * 256);
    v2f bv;
    bv.x = tb[ks * 512];
    bv.y = tb[(ks + 1) * 512];
    acc = wmma4(av, bv, acc);
  }
  #pragma unroll
  for (int v = 0; v < 8; ++v)
    OUT[obase + (r0 + v + 8 * hi) * 16 + m] = acc[v];
}

// ---------------------------------------------------------------------------
extern "C" void kernel_launch(void* const* d_in, const int* in_sizes, int n_in,
                              void* d_out, int out_size, void* d_ws, size_t ws_size,
                              hipStream_t stream) {
  (void)in_sizes; (void)n_in; (void)out_size; (void)ws_size;
  const float* X    = (const float*)d_in[0];   // (32,64,64,16)
  const float* W    = (const float*)d_in[1];   // (16,16,448,448)
  const float* Bias = (const float*)d_in[2];   // (16,16)
  float* OUT = (float*)d_out;                  // (32,64,64,16)

  float* ws = (float*)d_ws;
  float* Fp = ws;                              //     16384 floats (32x256x2)
  float* Gp = ws + 16384;                      //     16384 floats (128x64x2)
  float* T1 = ws + 32768;                      //  8,388,608 floats (reused as T2)
  float* XT = ws + 32768 + 8388608;            // 33,554,432 floats (also holds Y)
  // total workspace: ~160 MB

  swt_init_mats<<<1,     320, 0, stream>>>(Fp, Gp);
  swt_fwd_c    <<<4096,  256, 0, stream>>>(X, Fp, T1);
  swt_fwd_b    <<<16384, 256, 0, stream>>>(T1, Fp, XT);
  swt_mix      <<<8192,  256, 0, stream>>>(XT, W);          // in-place XT -> Y
  swt_skip     <<<1024,  256, 0, stream>>>(X, Bias, OUT);
  swt_inv_b    <<<4096,  256, 0, stream>>>(XT, Gp, T1);     // T1 reused as T2
  swt_inv_c    <<<1024,  256, 0, stream>>>(T1, Gp, OUT);
}